// PatchesToImage_82222853915002
// MI455X (gfx1250) — compile-verified
//
#include <hip/hip_runtime.h>
#include <cstdint>

// PatchesToImage: [64, 1024, 768] f32  ->  [64, 512, 512, 3] f32
// Pure permutation in 192-byte contiguous segments; HBM-bound (~402 MB total,
// floor ~17.3 us at 23.3 TB/s). Data path: CDNA5 async DMA
// global -> LDS -> global (b128), no VGPR staging of payload data.

namespace {
constexpr uint32_t kThreads      = 256;                       // 8 waves / block
constexpr uint32_t kVecPerThread = 8;                         // 8 x 16B each
constexpr uint32_t kVecPerBlock  = kThreads * kVecPerThread;  // 2048
constexpr uint32_t kTotalVec     = 64u * 1024u * 768u / 4u;   // 12,582,912 float4s
constexpr uint32_t kBlocks       = kTotalVec / kVecPerBlock;  // 6144 (exact)
constexpr uint32_t kLdsBytes     = kVecPerBlock * 16u;        // 32 KB
}

__global__ __launch_bounds__(kThreads) void patches_to_image_async(
    const float* __restrict__ in, float* __restrict__ out)
{
  extern __shared__ char smem[];

  // Low 32 bits of the generic (flat) pointer to LDS == byte offset within
  // the wave's LDS allocation (flat LDS aperture truncates to addr[31:0]).
  const uint32_t ldsBase = (uint32_t)(uintptr_t)(&smem[0]);

  const uint32_t tid  = threadIdx.x;
  const uint32_t v0   = blockIdx.x * kVecPerBlock + tid;
  const uint64_t inB  = (uint64_t)(uintptr_t)in;
  const uint64_t outB = (uint64_t)(uintptr_t)out;

  // ---- Phase 1: coalesced async loads, global (input-linear) -> LDS -------
  // Per instruction a wave pulls 32 x 16B = 512B contiguous.
#pragma unroll
  for (uint32_t k = 0; k < kVecPerThread; ++k) {
    const uint32_t v   = v0 + k * kThreads;
    const uint32_t lds = ldsBase + (k * kThreads + tid) * 16u;
    const uint64_t ga  = inB + (uint64_t)v * 16u;
    asm volatile("global_load_async_to_lds_b128 %0, %1, off"
                 :: "v"(lds), "v"(ga)
                 : "memory");
  }

  // Each thread only consumes LDS slots loaded by its own wave's async ops,
  // so a per-wave counter wait suffices (no workgroup barrier needed).
  asm volatile("s_wait_asynccnt 0x0" ::: "memory");

  // ---- Phase 2: async stores, LDS -> global (permuted, 192B bursts) -------
  // v enumerates input-linearly; decompose into (b, r, c, i, q):
  //   q   = v % 12   (float4 within the 48-float (j,ch) segment)
  //   seg = v / 12 = ((b*32 + r)*32 + c)*16 + i
  // Output float4 index: (b*512 + r*16 + i)*384 + c*12 + q.
#pragma unroll
  for (uint32_t k = 0; k < kVecPerThread; ++k) {
    const uint32_t v   = v0 + k * kThreads;
    const uint32_t q   = v % 12u;
    const uint32_t seg = v / 12u;
    const uint32_t i   = seg & 15u;
    const uint32_t c   = (seg >> 4) & 31u;
    const uint32_t r   = (seg >> 9) & 31u;
    const uint32_t b   = seg >> 14;
    const uint32_t ov  = (b * 512u + r * 16u + i) * 384u + c * 12u + q;

    const uint32_t lds = ldsBase + (k * kThreads + tid) * 16u;
    const uint64_t ga  = outB + (uint64_t)ov * 16u;
    asm volatile("global_store_async_from_lds_b128 %0, %1, off"
                 :: "v"(ga), "v"(lds)
                 : "memory");
  }

  // Drain outstanding async stores before the wave retires / LDS is reused.
  asm volatile("s_wait_asynccnt 0x0" ::: "memory");
}

extern "C" void kernel_launch(void* const* d_in, const int* in_sizes, int n_in,
                              void* d_out, int out_size, void* d_ws, size_t ws_size,
                              hipStream_t stream) {
  const float* in = (const float*)d_in[0];
  float* out      = (float*)d_out;
  patches_to_image_async<<<kBlocks, kThreads, kLdsBytes, stream>>>(in, out);
}